// GCN_5858335392232
// MI455X (gfx1250) — compile-verified
//
#include <hip/hip_runtime.h>

typedef float v2f __attribute__((ext_vector_type(2)));
typedef float v8f __attribute__((ext_vector_type(8)));
typedef float f4  __attribute__((ext_vector_type(4)));

#define FEAT 128

// ---------------------------------------------------------------- utilities
__global__ __launch_bounds__(256) void zero_f32(float* __restrict__ p, int n) {
    int i = blockIdx.x * 256 + threadIdx.x;
    if (i < n) p[i] = 0.0f;
}

// deg[dst] += 1 per edge (float so we can reuse f32 atomics)
__global__ __launch_bounds__(256) void deg_count(const long long* __restrict__ dst,
                                                 int E, float* __restrict__ deg) {
    int e = blockIdx.x * 256 + threadIdx.x;
    if (e < E) atomicAdd(&deg[(int)dst[e]], 1.0f);
}

// dinv = rsqrt(deg + 1)  (self loop folded into degree)
__global__ __launch_bounds__(256) void deg_inv(const float* __restrict__ deg,
                                               float* __restrict__ dinv, int n) {
    int i = blockIdx.x * 256 + threadIdx.x;
    if (i < n) dinv[i] = rsqrtf(deg[i] + 1.0f);
}

// ------------------------------------------------------------- WMMA GEMM
// H[n,128] = (relu_in ? relu(X) : X)[n,128] @ W[128,128]
// One block = 16 rows; 8 waves, wave w computes the 16-column tile w.
// fp32 WMMA 16x16x4, K-loop of 32 steps. A staged via LDS (pad 132 ->
// lanes 0..15 read 16 distinct banks), B read from W (64KB, L2-resident).
__global__ __launch_bounds__(256) void gemm128_wmma(const float* __restrict__ X,
                                                    const float* __restrict__ W,
                                                    float* __restrict__ H,
                                                    int n, int relu_in) {
    __shared__ float As[16][132];
    const int rowBase = blockIdx.x * 16;
    const int tid = threadIdx.x;

    // stage 16x128 tile of X into LDS, coalesced float4, fold previous ReLU
    {
        int base = tid * 8;                 // 256 threads * 8 floats = 2048
        #pragma unroll
        for (int j = 0; j < 8; j += 4) {
            int idx = base + j;
            int r = idx >> 7, c = idx & 127;
            f4 v = {0.f, 0.f, 0.f, 0.f};
            if (rowBase + r < n)
                v = *(const f4*)(X + (size_t)(rowBase + r) * FEAT + c);
            if (relu_in) {
                v.x = fmaxf(v.x, 0.f); v.y = fmaxf(v.y, 0.f);
                v.z = fmaxf(v.z, 0.f); v.w = fmaxf(v.w, 0.f);
            }
            *(f4*)&As[r][c] = v;            // row stride 528B -> 16B aligned
        }
    }
    __syncthreads();

    const int w     = tid >> 5;             // wave id -> column tile
    const int l     = tid & 31;             // lane
    const int m     = l & 15;               // row within tile / N within tile
    const int khalf = (l >> 4) << 1;        // lanes 16..31 hold K+2,K+3
    const int colB  = (w << 4) + m;         // output column

    v8f acc = {};
    #pragma unroll
    for (int k0 = 0; k0 < FEAT; k0 += 4) {
        int kk = k0 + khalf;
        v2f a, b;
        a.x = As[m][kk];
        a.y = As[m][kk + 1];
        b.x = W[(size_t)kk * FEAT + colB];
        b.y = W[(size_t)(kk + 1) * FEAT + colB];
        // v_wmma_f32_16x16x4_f32: D = A(16x4) * B(4x16) + C
        acc = __builtin_amdgcn_wmma_f32_16x16x4_f32(
            false, a, false, b, (short)0, acc, false, false);
    }

    // C layout: VGPR i -> M = i + 8*(l>>4), N = l&15
    #pragma unroll
    for (int i = 0; i < 8; ++i) {
        int M = i + ((l >> 4) << 3);
        if (rowBase + M < n)
            H[(size_t)(rowBase + M) * FEAT + colB] = acc[i];
    }
}

// -------------------------------------------------- self loop + bias init
// A[i,f] = H[i,f] * dinv[i]^2 + b[f]   (fully rewrites A before scatter)
__global__ __launch_bounds__(256) void self_bias(const float* __restrict__ H,
                                                 const float* __restrict__ dinv,
                                                 const float* __restrict__ bias,
                                                 float* __restrict__ A, int total) {
    int i = blockIdx.x * 256 + threadIdx.x;
    if (i < total) {
        int r = i >> 7, f = i & 127;
        float di = dinv[r];
        A[i] = H[i] * di * di + bias[f];
    }
}

// ------------------------------------------------------------- edge scatter
// One wave per edge: lane k handles floats [4k,4k+4) of the 128-wide row.
// 512B contiguous gather per edge (global_load_b128 per lane), then 4
// global_atomic_add_f32 per lane into agg[dst]. Both H and A are L2-resident
// (51.2 MB each << 192 MB L2), so this runs at L2 atomic rate.
__global__ __launch_bounds__(256) void edge_scatter(const float* __restrict__ H,
                                                    const float* __restrict__ dinv,
                                                    const long long* __restrict__ src,
                                                    const long long* __restrict__ dst,
                                                    float* __restrict__ A, int E) {
    int gid  = blockIdx.x * 256 + threadIdx.x;
    int e    = gid >> 5;
    int lane = gid & 31;
    if (e >= E) return;
    int s = (int)src[e];
    int d = (int)dst[e];
    float coef = dinv[s] * dinv[d];
    f4 h = *(const f4*)(H + (size_t)s * FEAT + lane * 4);
    float* out = A + (size_t)d * FEAT + lane * 4;
    atomicAdd(out + 0, h.x * coef);
    atomicAdd(out + 1, h.y * coef);
    atomicAdd(out + 2, h.z * coef);
    atomicAdd(out + 3, h.w * coef);
}

// ---------------------------------------------------------------- driver
extern "C" void kernel_launch(void* const* d_in, const int* in_sizes, int n_in,
                              void* d_out, int out_size, void* d_ws, size_t ws_size,
                              hipStream_t stream) {
    const float*     x  = (const float*)d_in[0];
    const long long* ei = (const long long*)d_in[1];   // int64 in reference
    const float* W1 = (const float*)d_in[2];
    const float* b1 = (const float*)d_in[3];
    const float* W2 = (const float*)d_in[4];
    const float* b2 = (const float*)d_in[5];
    const float* W3 = (const float*)d_in[6];
    const float* b3 = (const float*)d_in[7];

    const int n = in_sizes[0] / FEAT;
    const int E = in_sizes[1] / 2;
    const long long* src = ei;
    const long long* dst = ei + E;

    float* H    = (float*)d_ws;                 // [n,128] post-GEMM features
    float* deg  = H + (size_t)n * FEAT;         // [n]
    float* dinv = deg + n;                      // [n]
    float* A    = (float*)d_out;                // aggregation buffer, reused

    const int total = n * FEAT;
    dim3 blk(256);
    int gN   = (n + 255) / 256;
    int gE   = (E + 255) / 256;
    int gTot = (total + 255) / 256;
    int gGemm = (n + 15) / 16;
    int gScat = (int)(((long long)E * 32 + 255) / 256);

    // degree / normalization (recomputed every call: deterministic)
    zero_f32 <<<gN, blk, 0, stream>>>(deg, n);
    deg_count<<<gE, blk, 0, stream>>>(dst, E, deg);
    deg_inv  <<<gN, blk, 0, stream>>>(deg, dinv, n);

    // layer 1
    gemm128_wmma<<<gGemm, blk, 0, stream>>>(x, W1, H, n, 0);
    self_bias   <<<gTot, blk, 0, stream>>>(H, dinv, b1, A, total);
    edge_scatter<<<gScat, blk, 0, stream>>>(H, dinv, src, dst, A, E);

    // layer 2 (ReLU of layer-1 output folded into GEMM A-load)
    gemm128_wmma<<<gGemm, blk, 0, stream>>>(A, W2, H, n, 1);
    self_bias   <<<gTot, blk, 0, stream>>>(H, dinv, b2, A, total);
    edge_scatter<<<gScat, blk, 0, stream>>>(H, dinv, src, dst, A, E);

    // layer 3 (no final ReLU)
    gemm128_wmma<<<gGemm, blk, 0, stream>>>(A, W3, H, n, 1);
    self_bias   <<<gTot, blk, 0, stream>>>(H, dinv, b3, A, total);
    edge_scatter<<<gScat, blk, 0, stream>>>(H, dinv, src, dst, A, E);
}